// HOGLayerC_27625229647909
// MI455X (gfx1250) — compile-verified
//
#include <hip/hip_runtime.h>
#include <math.h>

// HOG layer for MI455X (gfx1250, wave32).
// Memory-bound: 462 MB output write => ~20us floor @ 23.3 TB/s.
// Bilinear 8x upsample done as out = A * P * A^T via V_WMMA_F32_16X16X4_F32
// (fp32 WMMA keeps exact fp32 accumulation, matching the reference).

typedef float v2f __attribute__((ext_vector_type(2)));
typedef float v8f __attribute__((ext_vector_type(8)));

#define HH 448
#define WW 448
#define NB 64          // batch
#define NBINS 9
#define PH 56          // pooled dim (448/8)
#define PPAD 64        // pooled dim padded to multiple of 16
#define NIMG (NB*NBINS) // 576 independent 56x56 -> 448x448 resizes

// Workspace layout (in floats):
#define A_OFF 0
#define A_SZ  (HH*PPAD)             // 448x64 interp matrix (cols >=56 are 0)
#define G_OFF (A_OFF + A_SZ)
#define G_SZ  256                   // 16x16 gaussian
#define P_OFF (G_OFF + G_SZ)
#define P_SZ  ((size_t)NIMG*PPAD*PPAD)   // padded pooled, zero-initialized
#define T_OFF (P_OFF + P_SZ)
#define T_SZ  ((size_t)NIMG*HH*PPAD)     // T = A*P intermediate

// ---------------------------------------------------------------- setup ----
// Blocks 0..111: A[j][k] = bilinear weight of pooled col k on out col j.
// jax.image.resize 'bilinear' upsample: src = (j+0.5)/8 - 0.5, edge clamped.
// Block 112: G = outer(w,w)/sum, w = exp(-0.5*((i-7.5)/8)^2).
__global__ void hog_setup(float* __restrict__ ws) {
    float* A = ws + A_OFF;
    float* G = ws + G_OFF;
    int t = threadIdx.x;
    if (blockIdx.x < (HH*PPAD)/256) {
        int idx = blockIdx.x * 256 + t;
        int j = idx / PPAD, k = idx % PPAD;
        float s  = j * 0.125f - 0.4375f;
        float ff = floorf(s);
        int   k0 = (int)ff;
        float w1 = s - ff;
        int c0 = min(max(k0,     0), PH - 1);
        int c1 = min(max(k0 + 1, 0), PH - 1);
        float a = 0.0f;
        if (c0 == k) a += 1.0f - w1;
        if (c1 == k) a += w1;
        A[idx] = (k < PH) ? a : 0.0f;
    } else {
        int u = t >> 4, v = t & 15;
        float S = 0.0f;
        for (int i = 0; i < 16; ++i) {
            float n = (i - 7.5f) * 0.125f;
            S += expf(-0.5f * n * n);
        }
        float nu = (u - 7.5f) * 0.125f;
        float nv = (v - 7.5f) * 0.125f;
        G[t] = expf(-0.5f * nu * nu) * expf(-0.5f * nv * nv) / (S * S);
    }
}

// ------------------------------------------------------------- HOG+pool ----
// One 256-thread block per 32x32 tile (4 px/thread): Sobel on reflect-padded
// input, mag*gauss scattered into LDS hist (4x4 cells x 9 bins, ds_add_f32),
// then 144 threads write the padded pooled buffer.
__global__ void hog_hist(const float* __restrict__ x, float* __restrict__ ws) {
    const float* G = ws + G_OFF;
    float*       P = ws + P_OFF;
    __shared__ float lh[16 * NBINS];
    int t = threadIdx.x;
    if (t < 16 * NBINS) lh[t] = 0.0f;
    __syncthreads();

    int b  = blockIdx.z;
    int ti = blockIdx.y, tj = blockIdx.x;
    const float* xb = x + (size_t)b * HH * WW;

    for (int p = 0; p < 4; ++p) {
        int idx = t + p * 256;
        int lr = idx >> 5, lc = idx & 31;
        int i = ti * 32 + lr, j = tj * 32 + lc;
        int im1 = (i == 0)      ? 1       : i - 1;   // reflect pad
        int ip1 = (i == HH - 1) ? HH - 2  : i + 1;
        int jm1 = (j == 0)      ? 1       : j - 1;
        int jp1 = (j == WW - 1) ? WW - 2  : j + 1;
        float x00 = xb[(size_t)im1*WW + jm1], x01 = xb[(size_t)im1*WW + j], x02 = xb[(size_t)im1*WW + jp1];
        float x10 = xb[(size_t)i  *WW + jm1],                               x12 = xb[(size_t)i  *WW + jp1];
        float x20 = xb[(size_t)ip1*WW + jm1], x21 = xb[(size_t)ip1*WW + j], x22 = xb[(size_t)ip1*WW + jp1];
        float gx = (x00 + 2.0f*x10 + x20) - (x02 + 2.0f*x12 + x22);
        float gy = (x00 + 2.0f*x01 + x02) - (x20 + 2.0f*x21 + x22);
        float mag = sqrtf(gx*gx + gy*gy) * G[(i & 15) * 16 + (j & 15)];
        float ph  = atan2f(gx, gy) * ((float)NBINS / 3.14159265358979323846f);
        int bin = (int)floorf(ph);
        bin = ((bin % NBINS) + NBINS) % NBINS;
        int cell = ((lr >> 3) << 2) + (lc >> 3);
        atomicAdd(&lh[cell * NBINS + bin], mag);
    }
    __syncthreads();

    if (t < 16 * NBINS) {
        int cell = t / NBINS, bin = t % NBINS;
        int pr = ti * 4 + (cell >> 2);
        int pc = tj * 4 + (cell & 3);
        P[((size_t)(b * NBINS + bin)) * PPAD * PPAD + pr * PPAD + pc] = lh[t];
    }
}

// ----------------------------------------------------------------- WMMA ----
__device__ __forceinline__ v8f wmma_f32x4(v2f a, v2f b, v8f c) {
    // 8 args: (neg_a, A, neg_b, B, c_mod, C, reuse_a, reuse_b)
    return __builtin_amdgcn_wmma_f32_16x16x4_f32(false, a, false, b,
                                                 (short)0, c, false, false);
}

// GEMM1: T[img](448x64) = A(448x64) * P[img](64x64). One wave per 16x16 tile,
// 4 waves/block cover the 4 N-tiles. K loop: 16 x (16x16x4) WMMAs.
__global__ void hog_gemm1(const float* __restrict__ ws_c, float* __restrict__ ws) {
    const float* A = ws_c + A_OFF;
    const float* P = ws_c + P_OFF + (size_t)blockIdx.y * (PPAD * PPAD);
    float*       T = ws   + T_OFF + (size_t)blockIdx.y * (HH * PPAD);
    int lane = threadIdx.x & 31;
    int wid  = threadIdx.x >> 5;           // n-tile 0..3
    int m0   = blockIdx.x * 16;
    int n0   = wid * 16;
    int row  = lane & 15;
    int kh   = (lane >> 4) * 2;            // lanes 16-31 hold K+2,K+3
    v8f c = {};
    for (int kk = 0; kk < PPAD; kk += 4) {
        v2f a = *(const v2f*)(A + (size_t)(m0 + row) * PPAD + kk + kh);
        v2f b;
        b.x = P[(size_t)(kk + kh + 0) * PPAD + n0 + row];
        b.y = P[(size_t)(kk + kh + 1) * PPAD + n0 + row];
        c = wmma_f32x4(a, b, c);
    }
    int mbase = m0 + (lane >> 4) * 8;
    for (int i = 0; i < 8; ++i)
        T[(size_t)(mbase + i) * PPAD + n0 + row] = c[i];
}

// GEMM2: out[img](448x448) = T(448x64) * A^T(64x448). B-frag of A^T is a
// contiguous float2 from row (n0+row) of A — both operands stream as B64.
__global__ void hog_gemm2(const float* __restrict__ ws, float* __restrict__ out) {
    const float* A = ws + A_OFF;
    const float* T = ws + T_OFF + (size_t)blockIdx.z * (HH * PPAD);
    float*       o = out + (size_t)blockIdx.z * HH * WW;
    int lane = threadIdx.x & 31;
    int wid  = threadIdx.x >> 5;
    int m0   = blockIdx.x * 16;
    int n0   = (blockIdx.y * 4 + wid) * 16;
    int row  = lane & 15;
    int kh   = (lane >> 4) * 2;
    v8f c = {};
    for (int kk = 0; kk < PPAD; kk += 4) {
        v2f a = *(const v2f*)(T + (size_t)(m0 + row) * PPAD + kk + kh);
        v2f b = *(const v2f*)(A + (size_t)(n0 + row) * PPAD + kk + kh);
        c = wmma_f32x4(a, b, c);
    }
    int mbase = m0 + (lane >> 4) * 8;
    for (int i = 0; i < 8; ++i)
        o[(size_t)(mbase + i) * WW + n0 + row] = c[i];
}

// --------------------------------------------------------------- launch ----
extern "C" void kernel_launch(void* const* d_in, const int* in_sizes, int n_in,
                              void* d_out, int out_size, void* d_ws, size_t ws_size,
                              hipStream_t stream) {
    (void)in_sizes; (void)n_in; (void)out_size; (void)ws_size;
    const float* x   = (const float*)d_in[0];
    float*       out = (float*)d_out;
    float*       ws  = (float*)d_ws;

    // Zero the padded pooled buffer (padding rows/cols must stay 0).
    hipMemsetAsync(ws + P_OFF, 0, P_SZ * sizeof(float), stream);

    hog_setup<<<dim3((HH*PPAD)/256 + 1), 256, 0, stream>>>(ws);
    hog_hist <<<dim3(WW/32, HH/32, NB), 256, 0, stream>>>(x, ws);
    hog_gemm1<<<dim3(HH/16, NIMG),        128, 0, stream>>>(ws, ws);
    hog_gemm2<<<dim3(HH/16, WW/64, NIMG), 128, 0, stream>>>(ws, out);
}